// NonLinearSAGE_54400055771176
// MI455X (gfx1250) — compile-verified
//
#include <hip/hip_runtime.h>

// NonLinearSAGE on MI455X (gfx1250, wave32).
// Pure scatter/stream workload: ~420 MB of traffic, ~10 FLOPs/output
// -> HBM/atomic bound (roofline ~18us @ 23.3 TB/s); matrix engines not
// applicable (every GEMM in the reference is a scalar multiply).
// Strategy:
//   * NT-stream edge_index (384 MB, one pass, 128-bit loads) so the stream
//     doesn't evict the L2-resident working set (x: 12 MB, agg: 4 MB, both
//     trivially fit in the 192 MB L2),
//   * global_prefetch_b8 one grid-stride ahead (speculative, OOB-safe),
//   * no-return global_atomic_add_f32 into a COMPACTED accumulator: the
//     reference only keeps h[::3], so 2/3 of all edges need no atomic at all,
//   * fused combine+slice+MLP epilogue, fully coalesced, NT output store.

#define N_NODES 3000000
#define N_EDGES 48000000
#define N_OUT   1000000   // N_NODES / 3

// Native Clang vectors: required by __builtin_nontemporal_load/store
typedef int   vint4   __attribute__((ext_vector_type(4)));
typedef float vfloat4 __attribute__((ext_vector_type(4)));

// --- hot kernel first so the disasm snippet shows it -----------------------
// Each thread consumes 4 edges per iteration via 128-bit NT loads of src/dst,
// prefetches its next chunk (global_prefetch_b8), and issues no-return
// global_atomic_add_f32 only for destinations that survive h[::3].
__global__ __launch_bounds__(256) void sage_scatter(const vint4* __restrict__ src4,
                                                    const vint4* __restrict__ dst4,
                                                    const float* __restrict__ x,
                                                    float* __restrict__ agg,
                                                    int n4) {
    const int stride = gridDim.x * blockDim.x;
    for (int i = blockIdx.x * blockDim.x + threadIdx.x; i < n4; i += stride) {
        // gfx1250 speculative prefetch of next iteration's edge chunks.
        // Speculative prefetches past the end of the buffer are dropped by
        // hardware, so no bounds check needed.
        __builtin_prefetch(&dst4[i + stride], 0, 0);
        __builtin_prefetch(&src4[i + stride], 0, 0);

        // Non-temporal 128-bit loads: 384 MB single-pass stream; keep L2 for x/agg.
        vint4 d = __builtin_nontemporal_load(&dst4[i]);
        vint4 s = __builtin_nontemporal_load(&src4[i]);

        unsigned d0 = (unsigned)d.x, d1 = (unsigned)d.y;
        unsigned d2 = (unsigned)d.z, d3 = (unsigned)d.w;
        if (d0 % 3u == 0u) atomicAdd(&agg[d0 / 3u], x[s.x]);
        if (d1 % 3u == 0u) atomicAdd(&agg[d1 / 3u], x[s.y]);
        if (d2 % 3u == 0u) atomicAdd(&agg[d2 / 3u], x[s.z]);
        if (d3 % 3u == 0u) atomicAdd(&agg[d3 / 3u], x[s.w]);
    }
}

__global__ __launch_bounds__(256) void sage_zero_agg(vfloat4* __restrict__ agg4, int n4) {
    int i = blockIdx.x * blockDim.x + threadIdx.x;
    if (i < n4) {
        vfloat4 z = {0.f, 0.f, 0.f, 0.f};
        agg4[i] = z;
    }
}

// 4 outputs/thread, fully coalesced: agg as float4, x[3i] picked out of three
// contiguous float4 loads (indices 12t+{0,3,6,9}). All weights are scalars ->
// the "SAGEConv + MLP" collapses to a handful of FMAs per output.
__global__ __launch_bounds__(256) void sage_epilogue(const vfloat4* __restrict__ agg4,
                                                     const vfloat4* __restrict__ x4,
                                                     const float* __restrict__ Wl,
                                                     const float* __restrict__ Wr,
                                                     const float* __restrict__ W1,
                                                     const float* __restrict__ b1,
                                                     const float* __restrict__ W2,
                                                     const float* __restrict__ b2,
                                                     vfloat4* __restrict__ out4,
                                                     int n4) {
    int t = blockIdx.x * blockDim.x + threadIdx.x;
    if (t >= n4) return;

    const float wl  = Wl[0], wr = Wr[0];
    const float w10 = W1[0], w11 = W1[1];
    const float c10 = b1[0], c11 = b1[1];
    const float w20 = W2[0], w21 = W2[1];
    const float c2  = b2[0];

    vfloat4 a  = agg4[t];
    vfloat4 xA = x4[3 * t + 0];   // x[12t .. 12t+3]
    vfloat4 xB = x4[3 * t + 1];   // x[12t+4 .. 12t+7]
    vfloat4 xC = x4[3 * t + 2];   // x[12t+8 .. 12t+11]
    float xs0 = xA.x;             // x[12t+0]
    float xs1 = xA.w;             // x[12t+3]
    float xs2 = xB.z;             // x[12t+6]
    float xs3 = xC.y;             // x[12t+9]

    float h0 = fmaf(a.x, wl, xs0 * wr);
    float h1 = fmaf(a.y, wl, xs1 * wr);
    float h2 = fmaf(a.z, wl, xs2 * wr);
    float h3 = fmaf(a.w, wl, xs3 * wr);

    vfloat4 o;
    o.x = fmaf(fmaxf(fmaf(h0, w10, c10), 0.f), w20,
          fmaf(fmaxf(fmaf(h0, w11, c11), 0.f), w21, c2));
    o.y = fmaf(fmaxf(fmaf(h1, w10, c10), 0.f), w20,
          fmaf(fmaxf(fmaf(h1, w11, c11), 0.f), w21, c2));
    o.z = fmaf(fmaxf(fmaf(h2, w10, c10), 0.f), w20,
          fmaf(fmaxf(fmaf(h2, w11, c11), 0.f), w21, c2));
    o.w = fmaf(fmaxf(fmaf(h3, w10, c10), 0.f), w20,
          fmaf(fmaxf(fmaf(h3, w11, c11), 0.f), w21, c2));

    // single-pass 4 MB output: non-temporal store
    __builtin_nontemporal_store(o, &out4[t]);
}

extern "C" void kernel_launch(void* const* d_in, const int* in_sizes, int n_in,
                              void* d_out, int out_size, void* d_ws, size_t ws_size,
                              hipStream_t stream) {
    const float* x   = (const float*)d_in[0];
    const int*   ei  = (const int*)d_in[1];      // [2, 48M] row-major int32
    const float* Wl  = (const float*)d_in[2];
    const float* Wr  = (const float*)d_in[3];
    const float* W1  = (const float*)d_in[4];
    const float* b1  = (const float*)d_in[5];
    const float* W2  = (const float*)d_in[6];
    const float* b2  = (const float*)d_in[7];
    float* out = (float*)d_out;
    float* agg = (float*)d_ws;                   // 1M floats = 4 MB (L2-resident)

    const vint4* src4 = (const vint4*)ei;
    const vint4* dst4 = (const vint4*)(ei + N_EDGES);

    // 1) zero compacted accumulator (d_ws is poisoned; must re-zero every call)
    {
        int n4 = N_OUT / 4;                      // 250,000 float4
        int blk = 256, grd = (n4 + blk - 1) / blk;
        sage_zero_agg<<<grd, blk, 0, stream>>>((vfloat4*)agg, n4);
    }
    // 2) edge scatter: 12M int4-chunks per array, grid-stride with prefetch
    {
        int n4 = N_EDGES / 4;                    // 12,000,000
        int blk = 256, grd = 4096;               // 1,048,576 threads, ~12 iters each
        sage_scatter<<<grd, blk, 0, stream>>>(src4, dst4, x, agg, n4);
    }
    // 3) fused SAGE combine + slice + MLP epilogue
    {
        int n4 = N_OUT / 4;                      // 250,000 threads, 4 outputs each
        int blk = 256, grd = (n4 + blk - 1) / blk;
        sage_epilogue<<<grd, blk, 0, stream>>>((const vfloat4*)agg, (const vfloat4*)x,
                                               Wl, Wr, W1, b1, W2, b2,
                                               (vfloat4*)out, n4);
    }
}